// SpikeAmplifier_73452530696745
// MI455X (gfx1250) — compile-verified
//
#include <hip/hip_runtime.h>
#include <cstdint>

// ---------------------------------------------------------------------------
// SpikeAmplifier IF-neuron scan for MI455X (gfx1250).
//
// Problem: T=128 sequential steps over 131,072 independent lanes, f32.
// Traffic = 64MiB in + 64MiB out  =>  ~5.8us floor @ 23.3 TB/s. Pure HBM-bound;
// no matrix math exists (nonlinear elementwise recurrence), so no WMMA.
// Strategy: 8-deep global_load_async_to_lds_b128 prefetch pipeline (ASYNCcnt)
// to maximize in-flight HBM bytes, VALU scan, nontemporal streaming stores.
// ---------------------------------------------------------------------------

typedef float v4f __attribute__((ext_vector_type(4)));

#define T_STEPS   128
#define N_BATCH   16
#define J_DIM     8192
#define S_STRIDE  (N_BATCH * J_DIM)      // 131072 f32 per timestep slice
#define BLOCK     256
#define VEC       4                      // floats per thread
#define DEPTH     8                      // async pipeline depth (power of 2)

__global__ __launch_bounds__(BLOCK)
void spike_amplifier_kernel(const float* __restrict__ in,
                            const float* __restrict__ lat_w,
                            float* __restrict__ out)
{
    // Per-thread 16B slots, DEPTH deep: 8 * 256 * 16B = 32 KB LDS (of 320 KB).
    __shared__ v4f buf[DEPTH][BLOCK];

    const int tid  = threadIdx.x;
    const int base = blockIdx.x * (BLOCK * VEC) + tid * VEC;  // index in N*J slice
    const int j    = base & (J_DIM - 1);                      // lateral-weight index

    const float* gsrc = in  + base;
    float*       gdst = out + base;

    // Low 32 bits of a generic LDS pointer == LDS byte offset (flat aperture rule).
    const uint32_t lds0      = (uint32_t)(uintptr_t)&buf[0][tid];
    const uint32_t ldsStride = (uint32_t)(BLOCK * sizeof(v4f));

    // ---- Prime the pipeline: DEPTH timesteps in flight (ASYNCcnt = DEPTH) ----
#pragma unroll
    for (int k = 0; k < DEPTH; ++k) {
        const uint64_t g = (uint64_t)(uintptr_t)(gsrc + (size_t)k * S_STRIDE);
        const uint32_t l = lds0 + (uint32_t)k * ldsStride;
        asm volatile("global_load_async_to_lds_b128 %0, %1, off"
                     :: "v"(l), "v"(g) : "memory");
    }

    // Lateral weight (constant per lane) and neuron state.
    v4f wv = *(const v4f*)(lat_w + j);
    v4f h  = {0.f, 0.f, 0.f, 0.f};
    v4f v  = {0.f, 0.f, 0.f, 0.f};
    v4f y  = {0.f, 0.f, 0.f, 0.f};

#pragma unroll 4
    for (int t = 0; t < T_STEPS; ++t) {
        // Oldest async load (timestep t) must have landed in LDS.
        asm volatile("s_wait_asynccnt %0" :: "n"(DEPTH - 1) : "memory");

        v4f x = buf[t & (DEPTH - 1)][tid];          // ds_load_b128

        // LDS read must complete before this slot is refilled by the async load.
        asm volatile("s_wait_dscnt 0" ::: "memory");

        // Refill the just-freed slot ((t+DEPTH) % DEPTH == t % DEPTH) with
        // timestep t+DEPTH; clamp address in the tail (dummy load, never read)
        // so the asynccnt arithmetic stays uniform.
        int tn = t + DEPTH;
        if (tn > T_STEPS - 1) tn = T_STEPS - 1;
        {
            const uint64_t g = (uint64_t)(uintptr_t)(gsrc + (size_t)tn * S_STRIDE);
            const uint32_t l = lds0 + (uint32_t)(t & (DEPTH - 1)) * ldsStride;
            asm volatile("global_load_async_to_lds_b128 %0, %1, off"
                         :: "v"(l), "v"(g) : "memory");
        }

        // IF neuron with lateral amplification:
        //   h = y_prev * (h + w); v += x + h; s = (v >= 1); v = s ? 0 : v
        h = y * (h + wv);
        v = v + (x + h);

        v4f s;
#pragma unroll
        for (int c = 0; c < VEC; ++c) {
            const bool fire = (v[c] >= 1.0f);
            s[c] = fire ? 1.0f : 0.0f;
            v[c] = fire ? 0.0f : v[c];
        }
        y = s;

        // Write-once streaming output: nontemporal store (TH=NT).
        __builtin_nontemporal_store(s, (v4f*)(gdst + (size_t)t * S_STRIDE));
    }
}

extern "C" void kernel_launch(void* const* d_in, const int* in_sizes, int n_in,
                              void* d_out, int out_size, void* d_ws, size_t ws_size,
                              hipStream_t stream) {
    const float* in    = (const float*)d_in[0];   // (T, N, C, J) f32
    const float* lat_w = (const float*)d_in[1];   // (J,) f32
    float*       out   = (float*)d_out;           // (T, N, C, J) f32

    const int lanes  = N_BATCH * J_DIM;           // 131072 independent lanes
    const int blocks = lanes / (BLOCK * VEC);     // 128 blocks * 256 threads
    spike_amplifier_kernel<<<blocks, BLOCK, 0, stream>>>(in, lat_w, out);
}